// Envelopes_12936441495779
// MI455X (gfx1250) — compile-verified
//
#include <hip/hip_runtime.h>
#include <cstdint>

typedef float v4f __attribute__((ext_vector_type(4)));

#define N_ITEMS 512
#define TS      2048
#define FULL    32768
#define PADDED  65536
#define NEVT    512      // BATCH * N_EVENTS

// ---------------- Phase 1: partial min/max of the item table ----------------
__global__ __launch_bounds__(256) void k_minmax_part(const float* __restrict__ items,
                                                     float* __restrict__ part) {
    __shared__ float smin[256], smax[256];
    const int tid = threadIdx.x;
    const int gid = blockIdx.x * 256 + tid;
    float mn = 3.402823e38f, mx = -3.402823e38f;
    const v4f* p = (const v4f*)items;
    for (int i = gid; i < (N_ITEMS * TS) / 4; i += 256 * 256) {
        v4f v = p[i];
        mn = fminf(mn, fminf(fminf(v.x, v.y), fminf(v.z, v.w)));
        mx = fmaxf(mx, fmaxf(fmaxf(v.x, v.y), fmaxf(v.z, v.w)));
    }
    smin[tid] = mn; smax[tid] = mx;
    __syncthreads();
    for (int s = 128; s > 0; s >>= 1) {
        if (tid < s) {
            smin[tid] = fminf(smin[tid], smin[tid + s]);
            smax[tid] = fmaxf(smax[tid], smax[tid + s]);
        }
        __syncthreads();
    }
    if (tid == 0) { part[blockIdx.x * 2] = smin[0]; part[blockIdx.x * 2 + 1] = smax[0]; }
}

// ---------------- Phase 2: final min + 1/(max-min+0.001) --------------------
__global__ __launch_bounds__(256) void k_minmax_final(float* __restrict__ ws) {
    __shared__ float smin[256], smax[256];
    const int tid = threadIdx.x;
    smin[tid] = ws[tid * 2];
    smax[tid] = ws[tid * 2 + 1];
    __syncthreads();
    for (int s = 128; s > 0; s >>= 1) {
        if (tid < s) {
            smin[tid] = fminf(smin[tid], smin[tid + s]);
            smax[tid] = fmaxf(smax[tid], smax[tid + s]);
        }
        __syncthreads();
    }
    if (tid == 0) {
        ws[512] = smin[0];                                   // global min
        ws[513] = 1.0f / ((smax[0] - smin[0]) + 0.001f);     // inv denom
    }
}

// ---------------- Phase 3: fused sparse-softmax + upsample + noise ----------
// One block (256 threads = 8 wave32) per (batch, event).
__global__ __launch_bounds__(256) void k_envelope(const float* __restrict__ selections,
                                                  const float* __restrict__ items,
                                                  const float* __restrict__ noise,
                                                  const float* __restrict__ stats,
                                                  float* __restrict__ out) {
    __shared__ float row[TS];     // raw selected item row (normalized at use via FMA)
    __shared__ float rv[256];
    __shared__ int   ri[256];

    const int be  = blockIdx.x;
    const int tid = threadIdx.x;

    // ---- argmax over 512 logits (first occurrence wins on ties) ----
    const float* sel = selections + be * N_ITEMS;
    const float v0 = sel[tid];
    const float v1 = sel[tid + 256];
    float mv; int mi;
    if (v1 > v0) { mv = v1; mi = tid + 256; } else { mv = v0; mi = tid; }
    rv[tid] = mv; ri[tid] = mi;
    __syncthreads();
    for (int s = 128; s > 0; s >>= 1) {
        if (tid < s) {
            const float ov = rv[tid + s]; const int oi = ri[tid + s];
            const float cv = rv[tid];     const int ci = ri[tid];
            if (ov > cv || (ov == cv && oi < ci)) { rv[tid] = ov; ri[tid] = oi; }
        }
        __syncthreads();
    }
    const float xmax = rv[0];
    const int   amax = __builtin_amdgcn_readfirstlane(ri[0]);
    __syncthreads();   // rv is reused below

    // ---- CDNA5 async DMA: stage selected row (8 KB) into LDS.
    // Issued now so it overlaps the exp-sum reduction below. GVS mode:
    // addr = SGPR64 base + per-lane VGPR byte offset; dest = per-lane LDS addr.
    {
        const uint64_t sbase = (uint64_t)(items + (size_t)amax * TS);  // uniform
        const unsigned lbase = (unsigned)(uintptr_t)(&row[0]);
        #pragma unroll
        for (int i = 0; i < 2; ++i) {
            const unsigned idx  = (unsigned)(tid + i * 256);   // 512 x b128 = 8192 B
            const unsigned goff = idx * 16u;
            const unsigned loff = lbase + idx * 16u;
            asm volatile("global_load_async_to_lds_b128 %0, %1, %2 offset:0"
                         :
                         : "v"(loff), "v"(goff), "s"(sbase)
                         : "memory");
        }
    }

    // ---- softmax denominator: w = max(softmax) = 1 / sum(exp(x - xmax)) ----
    rv[tid] = expf(v0 - xmax) + expf(v1 - xmax);
    __syncthreads();
    for (int s = 128; s > 0; s >>= 1) {
        if (tid < s) rv[tid] += rv[tid + s];
        __syncthreads();
    }
    const float w = 1.0f / rv[0];

    // Fold normalization + weight into one FMA: amp = raw*A + B
    const float mn     = stats[0];
    const float invden = stats[1];
    const float A = w * invden;
    const float B = -mn * A;

    // ---- wait for the async row DMA (own waves) + barrier (all waves) ----
    asm volatile("s_wait_asynccnt 0" ::: "memory");
    __syncthreads();

    // ---- 16x linear upsample * noise, streamed with NT b128 ----
    const v4f* nz = (const v4f*)(noise + (size_t)be * FULL);
    v4f*       ob = (v4f*)(out + (size_t)be * PADDED);
    for (int v = tid; v < FULL / 4; v += 256) {
        const v4f nv = __builtin_nontemporal_load(nz + v);
        const int j0 = v * 4;
        v4f amp;
        #pragma unroll
        for (int k = 0; k < 4; ++k) {
            float pos = ((float)(j0 + k) + 0.5f) * 0.0625f - 0.5f;
            pos = fminf(fmaxf(pos, 0.0f), 2047.0f);
            const int   lo = (int)pos;                 // pos >= 0 -> trunc == floor
            const float fr = pos - (float)lo;
            int hi = lo + 1; if (hi > TS - 1) hi = TS - 1;
            const float interp = row[lo] * (1.0f - fr) + row[hi] * fr;
            amp[k] = fmaf(interp, A, B);
        }
        __builtin_nontemporal_store(amp * nv, ob + v);
    }

    // ---- zero padding [FULL, PADDED) ----
    const v4f zero = {0.0f, 0.0f, 0.0f, 0.0f};
    v4f* zb = ob + FULL / 4;
    for (int v = tid; v < (PADDED - FULL) / 4; v += 256)
        __builtin_nontemporal_store(zero, zb + v);
}

extern "C" void kernel_launch(void* const* d_in, const int* in_sizes, int n_in,
                              void* d_out, int out_size, void* d_ws, size_t ws_size,
                              hipStream_t stream) {
    (void)in_sizes; (void)n_in; (void)out_size; (void)ws_size;
    const float* selections = (const float*)d_in[0];   // [8,64,512]
    const float* items      = (const float*)d_in[1];   // [512,2048]
    const float* noise      = (const float*)d_in[2];   // [8,64,32768]
    float*       out        = (float*)d_out;           // [8,64,65536]
    float*       ws         = (float*)d_ws;            // [0..511] partials, [512..513] stats

    k_minmax_part <<<256, 256, 0, stream>>>(items, ws);
    k_minmax_final<<<1,   256, 0, stream>>>(ws);
    k_envelope    <<<NEVT, 256, 0, stream>>>(selections, items, noise, ws + 512, out);
}